// GNN_imp_estimator_26362509263437
// MI455X (gfx1250) — compile-verified
//
#include <hip/hip_runtime.h>
#include <hip/hip_bf16.h>
#include <math.h>

// ---------------------------------------------------------------------------
// GNN importance estimator for MI455X (gfx1250, wave32, WMMA).
// Strategy: edge scatter/atomics dominate (memory bound); dense node matmuls
// use V_WMMA_F32_16X16X4_F32 (16x16 tile per wave, K stepped by 4).
// ---------------------------------------------------------------------------

typedef float v2f __attribute__((ext_vector_type(2)));
typedef float v8f __attribute__((ext_vector_type(8)));

#define NN_NODES 100000
#define NN_EDGES 1600000
#define NN_GRAPHS 256
#define BN_EPS 1e-5f

// ---------------------------- utility kernels ------------------------------

__global__ void fill_f32_kernel(float* __restrict__ p, float v, int n) {
    for (int i = blockIdx.x * blockDim.x + threadIdx.x; i < n;
         i += gridDim.x * blockDim.x)
        p[i] = v;
}

__global__ void degree_kernel(const int* __restrict__ src,
                              const int* __restrict__ dst,
                              float* __restrict__ odeg,
                              float* __restrict__ ideg, int E) {
    int i = blockIdx.x * blockDim.x + threadIdx.x;
    if (i >= E) return;
    atomicAdd(&odeg[src[i]], 1.0f);
    atomicAdd(&ideg[dst[i]], 1.0f);
}

__global__ void finalize_deg_kernel(float* __restrict__ d, int n) {
    int i = blockIdx.x * blockDim.x + threadIdx.x;
    if (i >= n) return;
    d[i] = rsqrtf(fmaxf(d[i], 1.0f));
}

// ------------------------- edge message aggregation ------------------------
// agg[dst] += h[src] * ns[src] * ew  (F floats per edge, 4 per thread)
template <int F>
__global__ void edge_aggregate_kernel(const float* __restrict__ h,
                                      const int* __restrict__ src,
                                      const int* __restrict__ dst,
                                      const float* __restrict__ ew,
                                      const float* __restrict__ ns,
                                      float* __restrict__ agg, int E) {
    constexpr int G = F / 4;  // float4 groups per edge
    int t = blockIdx.x * blockDim.x + threadIdx.x;
    int e = t / G;
    int q = t - e * G;
    if (e >= E) return;
    int s = src[e];
    int d = dst[e];
    float c = ns[s] * ew[e];
    const float4 v =
        *reinterpret_cast<const float4*>(h + (size_t)s * F + q * 4);
    float* o = agg + (size_t)d * F + q * 4;
    atomicAdd(o + 0, v.x * c);
    atomicAdd(o + 1, v.y * c);
    atomicAdd(o + 2, v.z * c);
    atomicAdd(o + 3, v.w * c);
}

// ------------------------------ WMMA matmul --------------------------------
// Y[M x N] = (A[M x K] * rowscale[M]) @ W[K x N] + bias[N]
// One wave computes one 16x16 output tile via V_WMMA_F32_16X16X4_F32.
// A layout (ISA 16x4 f32 table): lanes 0-15 hold M=0..15 / K={k0,k0+1},
// lanes 16-31 hold M=0..15 / K={k0+2,k0+3}.  B mirrors the K split with
// columns across lanes; C/D: VGPR r -> row r (+8 for upper lanes), col lane.
// Column tail (N=8) is handled branchlessly: clamp address, mask value.
template <int K, int N>
__global__ void wmma_matmul_kernel(const float* __restrict__ A,
                                   const float* __restrict__ W,
                                   const float* __restrict__ bias,
                                   const float* __restrict__ rowscale,
                                   float* __restrict__ Y, int M) {
    constexpr int NT = (N + 15) / 16;
    constexpr bool FULL = (N % 16) == 0;  // no column tail
    int lane = threadIdx.x & 31;
    int wave = blockIdx.x * (blockDim.x >> 5) + (threadIdx.x >> 5);
    int mtiles = M >> 4;  // M is a multiple of 16
    if (wave >= mtiles * NT) return;  // wave-uniform branch: EXEC stays all-1
    int mt = wave / NT;
    int nt = wave - mt * NT;
    int half = lane >> 4;
    int lr = lane & 15;
    int row = (mt << 4) + lr;
    int col = (nt << 4) + lr;
    bool cok = FULL || (col < N);
    int ccol = FULL ? col : (cok ? col : 0);  // always-in-bounds load column
    float sc = rowscale ? rowscale[row] : 1.0f;

    v8f c = {};
#pragma unroll
    for (int k0 = 0; k0 < K; k0 += 4) {
        int ka = k0 + 2 * half;
        v2f a, b;
        a.x = A[(size_t)row * K + ka] * sc;
        a.y = A[(size_t)row * K + ka + 1] * sc;
        float wx = W[ka * N + ccol];
        float wy = W[(ka + 1) * N + ccol];
        b.x = cok ? wx : 0.0f;
        b.y = cok ? wy : 0.0f;
        c = __builtin_amdgcn_wmma_f32_16x16x4_f32(
            /*neg_a=*/false, a, /*neg_b=*/false, b,
            /*c_mod=*/(short)0, c, /*reuse_a=*/false, /*reuse_b=*/false);
    }
    float bb = 0.0f;
    if (bias != nullptr) {
        float bv = bias[ccol];
        bb = cok ? bv : 0.0f;
    }
#pragma unroll
    for (int r = 0; r < 8; ++r) {
        int orow = (mt << 4) + r + 8 * half;
        if (cok) Y[(size_t)orow * N + col] = c[r] + bb;
    }
}

// --------------------------- batch-norm pieces -----------------------------

template <int N>
__global__ void colstats_kernel(const float* __restrict__ y, int total,
                                float* __restrict__ sums,
                                float* __restrict__ sumsq) {
    __shared__ float ss[32], sq[32];
    if (threadIdx.x < 32) {
        ss[threadIdx.x] = 0.0f;
        sq[threadIdx.x] = 0.0f;
    }
    __syncthreads();
    int idx0 = blockIdx.x * blockDim.x + threadIdx.x;
    int stride = gridDim.x * blockDim.x;  // multiple of N (N | 256)
    int col = idx0 & (N - 1);             // invariant across the loop
    float s = 0.0f, q = 0.0f;
    for (int i = idx0; i < total; i += stride) {
        float v = y[i];
        s += v;
        q += v * v;
    }
    atomicAdd(&ss[col], s);
    atomicAdd(&sq[col], q);
    __syncthreads();
    if (threadIdx.x < N) {
        atomicAdd(&sums[threadIdx.x], ss[threadIdx.x]);
        atomicAdd(&sumsq[threadIdx.x], sq[threadIdx.x]);
    }
}

__global__ void bn_finalize_kernel(const float* __restrict__ sums,
                                   const float* __restrict__ sumsq,
                                   const float* __restrict__ g,
                                   const float* __restrict__ be,
                                   float* __restrict__ scale,
                                   float* __restrict__ shift, int N,
                                   float invM) {
    int c = threadIdx.x;
    if (c >= N) return;
    float mu = sums[c] * invM;
    float var = fmaxf(sumsq[c] * invM - mu * mu, 0.0f);
    float inv = rsqrtf(var + BN_EPS);
    float sc = g[c] * inv;
    scale[c] = sc;
    shift[c] = be[c] - mu * sc;
}

__global__ void bn_apply_kernel(float* __restrict__ y,
                                const float* __restrict__ scale,
                                const float* __restrict__ shift, int total,
                                int nmask, int do_relu) {
    for (int i = blockIdx.x * blockDim.x + threadIdx.x; i < total;
         i += gridDim.x * blockDim.x) {
        int c = i & nmask;
        float v = y[i] * scale[c] + shift[c];
        if (do_relu) v = fmaxf(v, 0.0f);
        y[i] = v;
    }
}

// y[i,j] = y[i,j] * nd[i] + bias[j]   (post-aggregation normalize + bias)
template <int N>
__global__ void scale_bias_kernel(float* __restrict__ y,
                                  const float* __restrict__ nd,
                                  const float* __restrict__ bias, int M) {
    int i = blockIdx.x * blockDim.x + threadIdx.x;
    if (i >= M * N) return;
    int row = i / N;
    int col = i & (N - 1);
    y[i] = y[i] * nd[row] + bias[col];
}

// ----------------------------- readout head --------------------------------

__global__ void final_linear_kernel(const float* __restrict__ h,
                                    const float* __restrict__ Wl,
                                    const float* __restrict__ bl,
                                    float* __restrict__ s, int M) {
    int i = blockIdx.x * blockDim.x + threadIdx.x;
    if (i >= M) return;
    float acc = bl[0];
#pragma unroll
    for (int j = 0; j < 8; ++j) acc += h[(size_t)i * 8 + j] * Wl[j];
    s[i] = acc;
}

__device__ __forceinline__ unsigned enc_f32(float f) {
    unsigned u = __float_as_uint(f);
    return (u & 0x80000000u) ? ~u : (u | 0x80000000u);
}
__device__ __forceinline__ float dec_f32(unsigned u) {
    return __uint_as_float((u & 0x80000000u) ? (u ^ 0x80000000u) : ~u);
}

__global__ void seg_max_kernel(const float* __restrict__ s,
                               const int* __restrict__ batch,
                               unsigned* __restrict__ mx, int M) {
    int i = blockIdx.x * blockDim.x + threadIdx.x;
    if (i >= M) return;
    atomicMax(&mx[batch[i]], enc_f32(s[i]));
}

__global__ void exp_kernel(const float* __restrict__ s,
                           const int* __restrict__ batch,
                           const unsigned* __restrict__ mx,
                           float* __restrict__ e, float* __restrict__ den,
                           int M) {
    int i = blockIdx.x * blockDim.x + threadIdx.x;
    if (i >= M) return;
    int b = batch[i];
    float v = expf(s[i] - dec_f32(mx[b]));
    e[i] = v;
    atomicAdd(&den[b], v);
}

__global__ void normalize_kernel(float* __restrict__ e,
                                 const int* __restrict__ batch,
                                 const float* __restrict__ den, int M) {
    int i = blockIdx.x * blockDim.x + threadIdx.x;
    if (i >= M) return;
    e[i] /= den[batch[i]];
}

// ------------------------------- launcher ----------------------------------

extern "C" void kernel_launch(void* const* d_in, const int* in_sizes, int n_in,
                              void* d_out, int out_size, void* d_ws,
                              size_t ws_size, hipStream_t stream) {
    (void)in_sizes; (void)n_in; (void)out_size; (void)ws_size;

    const float* x   = (const float*)d_in[0];
    const float* ew  = (const float*)d_in[1];
    const float* W0  = (const float*)d_in[2];
    const float* b0  = (const float*)d_in[3];
    const float* g0  = (const float*)d_in[4];
    const float* be0 = (const float*)d_in[5];
    const float* W1  = (const float*)d_in[6];
    const float* b1  = (const float*)d_in[7];
    const float* g1  = (const float*)d_in[8];
    const float* be1 = (const float*)d_in[9];
    const float* W2  = (const float*)d_in[10];
    const float* b2  = (const float*)d_in[11];
    const float* g2  = (const float*)d_in[12];
    const float* be2 = (const float*)d_in[13];
    const float* Wl  = (const float*)d_in[14];
    const float* bl  = (const float*)d_in[15];
    const int* ei    = (const int*)d_in[16];
    const int* batch = (const int*)d_in[17];
    const int* srcI = ei;
    const int* dstI = ei + NN_EDGES;
    float* out = (float*)d_out;

    const int M = NN_NODES, E = NN_EDGES;

    // workspace layout (floats)
    float* ws    = (float*)d_ws;
    float* ns    = ws + 0;               // 100000  rsqrt(out_deg)
    float* nd    = ws + 100000;          // 100000  rsqrt(in_deg)
    float* bufA  = ws + 200000;          // 3.2M    y0 / h0 (100k x 32)
    float* bufB  = ws + 3400000;         // 1.6M    agg0, later p1, later p2
    float* bufC  = ws + 5000000;         // 1.6M    agg1 / h1 (100k x 16)
    float* bufD  = ws + 6600000;         // 0.8M    agg2 / h2 (100k x 8)
    float* sbuf  = ws + 7400000;         // 100000  node scores
    float* sums  = ws + 7500000;         // 32
    float* sumsq = ws + 7500032;         // 32
    float* scale = ws + 7500064;         // 32
    float* shift = ws + 7500096;         // 32
    unsigned* mx = (unsigned*)(ws + 7500128);  // 256 (0u == encoded "-inf")
    float* den   = ws + 7500384;         // 256
    // total: 7,500,640 floats ~= 30 MB

    const float invM = 1.0f / (float)M;
    const int T = 256;

    // --- zero only what must accumulate (ns/nd, agg regions, stats, mx, den)
    fill_f32_kernel<<<512, T, 0, stream>>>(ns, 0.0f, 200000);
    fill_f32_kernel<<<2048, T, 0, stream>>>(bufB, 0.0f, 4000000);  // B,C,D
    fill_f32_kernel<<<1, T, 0, stream>>>(sums, 0.0f, 640);  // stats..den

    // --- degree normalizers
    degree_kernel<<<(E + T - 1) / T, T, 0, stream>>>(srcI, dstI, ns, nd, E);
    finalize_deg_kernel<<<(M + T - 1) / T, T, 0, stream>>>(ns, M);
    finalize_deg_kernel<<<(M + T - 1) / T, T, 0, stream>>>(nd, M);

    // ------------------- layer 0: aggregate(x,16) then @W0(16x32) ----------
    edge_aggregate_kernel<16><<<(E * 4 + T - 1) / T, T, 0, stream>>>(
        x, srcI, dstI, ew, ns, bufB, E);
    {  // y0 = (agg0 * nd) @ W0 + b0  -> bufA
        int waves = (M / 16) * 2;
        wmma_matmul_kernel<16, 32><<<(waves * 32 + T - 1) / T, T, 0, stream>>>(
            bufB, W0, b0, nd, bufA, M);
    }
    colstats_kernel<32><<<1024, T, 0, stream>>>(bufA, M * 32, sums, sumsq);
    bn_finalize_kernel<<<1, 32, 0, stream>>>(sums, sumsq, g0, be0, scale,
                                             shift, 32, invM);
    bn_apply_kernel<<<2048, T, 0, stream>>>(bufA, scale, shift, M * 32, 31, 1);

    // ------------------- layer 1: h0 @ W1 first (32->16), then aggregate ---
    fill_f32_kernel<<<1, 64, 0, stream>>>(sums, 0.0f, 64);
    {  // p1 = h0 @ W1  -> bufB (overwrites agg0)
        int waves = (M / 16) * 1;
        wmma_matmul_kernel<32, 16><<<(waves * 32 + T - 1) / T, T, 0, stream>>>(
            bufA, W1, nullptr, nullptr, bufB, M);
    }
    edge_aggregate_kernel<16><<<(E * 4 + T - 1) / T, T, 0, stream>>>(
        bufB, srcI, dstI, ew, ns, bufC, E);
    scale_bias_kernel<16><<<(M * 16 + T - 1) / T, T, 0, stream>>>(bufC, nd,
                                                                  b1, M);
    colstats_kernel<16><<<1024, T, 0, stream>>>(bufC, M * 16, sums, sumsq);
    bn_finalize_kernel<<<1, 32, 0, stream>>>(sums, sumsq, g1, be1, scale,
                                             shift, 16, invM);
    bn_apply_kernel<<<2048, T, 0, stream>>>(bufC, scale, shift, M * 16, 15, 1);

    // ------------------- layer 2: h1 @ W2 first (16->8), then aggregate ----
    fill_f32_kernel<<<1, 64, 0, stream>>>(sums, 0.0f, 64);
    {  // p2 = h1 @ W2  -> bufB (first 800k floats)
        int waves = (M / 16) * 1;
        wmma_matmul_kernel<16, 8><<<(waves * 32 + T - 1) / T, T, 0, stream>>>(
            bufC, W2, nullptr, nullptr, bufB, M);
    }
    edge_aggregate_kernel<8><<<(E * 2 + T - 1) / T, T, 0, stream>>>(
        bufB, srcI, dstI, ew, ns, bufD, E);
    scale_bias_kernel<8><<<(M * 8 + T - 1) / T, T, 0, stream>>>(bufD, nd, b2,
                                                                M);
    colstats_kernel<8><<<1024, T, 0, stream>>>(bufD, M * 8, sums, sumsq);
    bn_finalize_kernel<<<1, 32, 0, stream>>>(sums, sumsq, g2, be2, scale,
                                             shift, 8, invM);
    bn_apply_kernel<<<2048, T, 0, stream>>>(bufD, scale, shift, M * 8, 7, 0);

    // ------------------- readout: linear + per-graph softmax ---------------
    final_linear_kernel<<<(M + T - 1) / T, T, 0, stream>>>(bufD, Wl, bl, sbuf,
                                                           M);
    seg_max_kernel<<<(M + T - 1) / T, T, 0, stream>>>(sbuf, batch, mx, M);
    exp_kernel<<<(M + T - 1) / T, T, 0, stream>>>(sbuf, batch, mx, out, den,
                                                  M);
    normalize_kernel<<<(M + T - 1) / T, T, 0, stream>>>(out, batch, den, M);
}